// MiniBatchDiscrimination_79113297592747
// MI455X (gfx1250) — compile-verified
//
#include <hip/hip_runtime.h>

// ---------------------------------------------------------------------------
// Problem constants (from the reference)
// ---------------------------------------------------------------------------
constexpr int BATCH = 512;
constexpr int IN_F  = 1024;
constexpr int OUT_F = 64;
constexpr int KDIM  = 32;                 // inter dim
constexpr int NCOL  = OUT_F * KDIM;       // 2048, flattened GEMM N
constexpr int OUT_W = IN_F + OUT_F;       // 1088, output row width

typedef __attribute__((ext_vector_type(2))) float v2f;
typedef __attribute__((ext_vector_type(8))) float v8f;

// ---------------------------------------------------------------------------
// Kernel 1: M[512,2048] = x[512,1024] @ W[1024,2048] using f32 WMMA 16x16x4.
// One wave (32 lanes) per 16x16 C tile; 256 WMMAs over K=1024.
// ---------------------------------------------------------------------------
__global__ __launch_bounds__(32)
void gemm_wmma(const float* __restrict__ x,
               const float* __restrict__ W,
               float* __restrict__ Mout)
{
    const int lane = threadIdx.x & 31;
    const int tile = blockIdx.x;            // 32 * 128 = 4096 tiles
    const int tn   = tile & 127;            // N tile index (2048/16)
    const int tm   = tile >> 7;             // M tile index (512/16)
    const int m0   = tm * 16;
    const int n0   = tn * 16;

    const int hi  = lane >> 4;              // 0: lanes 0-15, 1: lanes 16-31
    const int l16 = lane & 15;

    // A fragment (16x4, MxK): lane holds x[m0+l16][k + 2*hi + {0,1}] (b64 load)
    const float* aptr = x + (m0 + l16) * IN_F + 2 * hi;
    // B fragment (4x16, KxN): VGPR r holds W[k + 2*hi + r][n0 + l16]
    const float* bptr = W + (2 * hi) * NCOL + (n0 + l16);

    v8f acc = {};
#pragma unroll 4
    for (int k = 0; k < IN_F; k += 4) {
        v2f a = *(const v2f*)(aptr + k);    // two consecutive K values
        v2f b;
        b.x = bptr[k * NCOL];
        b.y = bptr[k * NCOL + NCOL];
        acc = __builtin_amdgcn_wmma_f32_16x16x4_f32(
            /*neg_a=*/false, a, /*neg_b=*/false, b,
            /*c_mod=*/(short)0, acc, /*reuse_a=*/false, /*reuse_b=*/false);
    }

    // C/D layout: VGPR v -> row (v + 8*hi), col l16
    float* cptr = Mout + (size_t)(m0 + 8 * hi) * NCOL + n0 + l16;
#pragma unroll
    for (int v = 0; v < 8; ++v)
        cptr[v * NCOL] = acc[v];
}

// ---------------------------------------------------------------------------
// Kernel 2: copy x into out[:, :1024] (float4, fully coalesced)
// ---------------------------------------------------------------------------
__global__ __launch_bounds__(256)
void copy_x(const float* __restrict__ x, float* __restrict__ out)
{
    int idx = blockIdx.x * 256 + threadIdx.x;   // 131072 float4 total
    int row = idx >> 8;                         // 256 float4 per input row
    int c   = idx & 255;
    *(float4*)(out + (size_t)row * OUT_W + c * 4) =
        *(const float4*)(x + (size_t)row * IN_F + c * 4);
}

// ---------------------------------------------------------------------------
// Kernel 3: minibatch features.
// Thread t owns (i, o); M[i,o,0:32] lives in 32 VGPRs. j streamed through LDS
// in 32-row tiles (broadcast reads). 2 VALU per k (v_sub + v_add with |mod|),
// one v_exp_f32 per j.
// ---------------------------------------------------------------------------
__global__ __launch_bounds__(256)
void pairwise(const float* __restrict__ Mmat,   // [512][2048] = [512][64][32]
              float* __restrict__ out)          // [512][1088]
{
    __shared__ float tile[32][KDIM];            // 4 KB

    const int o = blockIdx.x & 63;
    const int i = (blockIdx.x >> 6) * 256 + threadIdx.x;

    float mi[KDIM];
    const float* mrow = Mmat + (size_t)i * NCOL + o * KDIM;
#pragma unroll
    for (int k = 0; k < KDIM; k += 4) {
        float4 v = *(const float4*)(mrow + k);
        mi[k] = v.x; mi[k + 1] = v.y; mi[k + 2] = v.z; mi[k + 3] = v.w;
    }

    float acc = 0.0f;
    for (int j0 = 0; j0 < BATCH; j0 += 32) {
        // 32 rows x 32 floats = 1024 floats; 256 threads x float4
        {
            int jj = threadIdx.x >> 3;          // 0..31
            int kk = (threadIdx.x & 7) * 4;     // 0..28
            *(float4*)&tile[jj][kk] =
                *(const float4*)(Mmat + (size_t)(j0 + jj) * NCOL + o * KDIM + kk);
        }
        __syncthreads();

#pragma unroll 4
        for (int jj = 0; jj < 32; ++jj) {
            float s = 0.0f;
#pragma unroll
            for (int k = 0; k < KDIM; k += 4) {
                float4 t = *(const float4*)&tile[jj][k];
                s += fabsf(t.x - mi[k]);
                s += fabsf(t.y - mi[k + 1]);
                s += fabsf(t.z - mi[k + 2]);
                s += fabsf(t.w - mi[k + 3]);
            }
            acc += __expf(-s);                  // includes self term exp(0)=1
        }
        __syncthreads();
    }

    out[(size_t)i * OUT_W + IN_F + o] = acc;
}

// ---------------------------------------------------------------------------
// Launch
// ---------------------------------------------------------------------------
extern "C" void kernel_launch(void* const* d_in, const int* in_sizes, int n_in,
                              void* d_out, int out_size, void* d_ws, size_t ws_size,
                              hipStream_t stream)
{
    const float* x = (const float*)d_in[0];   // [512,1024]
    const float* W = (const float*)d_in[1];   // T reshaped [1024,2048] (contiguous)
    float* out = (float*)d_out;               // [512,1088]
    float* Mws = (float*)d_ws;                // [512,2048] scratch (4 MB)

    // GEMM: 32 x 128 = 4096 tiles, one wave each
    gemm_wmma<<<dim3(4096), dim3(32), 0, stream>>>(x, W, Mws);
    // x passthrough: 512*1024 floats / 4 / 256 = 512 blocks
    copy_x<<<dim3(512), dim3(256), 0, stream>>>(x, out);
    // features: 64 o * 2 i-tiles = 128 blocks
    pairwise<<<dim3(128), dim3(256), 0, stream>>>(Mws, out);
}